// DeChunkLayer_55319178773124
// MI455X (gfx1250) — compile-verified
//
#include <hip/hip_runtime.h>
#include <hip/hip_bf16.h>
#include <stdint.h>

// Problem constants (match reference).
#define BB   4
#define LL   4096
#define DD   1024
#define CC   32            // chunks per batch for the parallel scan
#define SS   (LL / CC)     // 128 scan steps per chunk
#define NBUF 4             // async LDS ring-buffer depth (power of 2)
#define EPSF 1e-4f

typedef float v4f __attribute__((ext_vector_type(4)));

// ---- CDNA5 async global->LDS copy (16B per lane) + split wait counters ----
__device__ __forceinline__ void async_copy16(uint32_t lds_addr, const float* g) {
  asm volatile("global_load_async_to_lds_b128 %0, %1, off"
               :
               : "v"(lds_addr), "v"((unsigned long long)(uintptr_t)g)
               : "memory");
}
// Wait until the oldest of `rem` outstanding row-copies has landed.
__device__ __forceinline__ void wait_async_rows(int rem) {
  if (rem > 3)       asm volatile("s_wait_asynccnt 0x3" ::: "memory");
  else if (rem == 3) asm volatile("s_wait_asynccnt 0x2" ::: "memory");
  else if (rem == 2) asm volatile("s_wait_asynccnt 0x1" ::: "memory");
  else               asm volatile("s_wait_asynccnt 0x0" ::: "memory");
}
__device__ __forceinline__ void wait_ds0() {
  asm volatile("s_wait_dscnt 0x0" ::: "memory");
}

// ---------------- K1: per-batch boundary compaction ----------------
// Writes: pScan[b][k] = clipped p of k-th boundary, bpos[b][k] = its position,
//         bpos[b][n_b] = L sentinel, nB[b] = n_b.
__global__ __launch_bounds__(1024)
void k1_prep(const int* __restrict__ bmask, const float* __restrict__ bprob,
             float* __restrict__ pScan, int* __restrict__ bpos,
             int* __restrict__ nB) {
  const int b = blockIdx.x;
  const int t = threadIdx.x;            // 0..1023
  __shared__ int sc[1024];
  __shared__ int s_base;
  if (t == 0) s_base = 0;
  __syncthreads();
  for (int tile = 0; tile < LL / 1024; ++tile) {
    const int l = tile * 1024 + t;
    const int m = (bmask[(size_t)b * LL + l] != 0) ? 1 : 0;
    sc[t] = m;
    __syncthreads();
    // Hillis-Steele inclusive scan over 1024 lanes
    for (int off = 1; off < 1024; off <<= 1) {
      int v = sc[t];
      if (t >= off) v += sc[t - off];
      __syncthreads();
      sc[t] = v;
      __syncthreads();
    }
    const int base = s_base;
    const int incl = sc[t];
    if (m) {
      const int idx = base + (incl - m);
      bpos[(size_t)b * (LL + 1) + idx] = l;
      float p = bprob[((size_t)b * LL + l) * 2 + 1];
      p = fminf(fmaxf(p, EPSF), 1.0f - EPSF);
      pScan[(size_t)b * LL + idx] = p;
    }
    __syncthreads();
    if (t == 1023) s_base = base + incl;
    __syncthreads();
  }
  if (t == 0) {
    const int tot = s_base;
    nB[b] = tot;
    bpos[(size_t)b * (LL + 1) + tot] = LL;  // sentinel
  }
}

// ---------------- K2: pass 1 — per-chunk local scan ----------------
// E[b,c,:] = chunk-local EMA end state (zero init), A[b,c] = prod(1-p) over chunk.
__global__ __launch_bounds__(256)
void k2_pass1(const float* __restrict__ hidden, const float* __restrict__ pScan,
              const int* __restrict__ nB, float* __restrict__ E,
              float* __restrict__ A) {
  const int c = blockIdx.x, b = blockIdx.y;
  const int t = threadIdx.x;            // 0..255; thread owns 4 floats of D
  __shared__ v4f  xbuf[NBUF][256];      // 4 rows x 4KB ring buffer
  __shared__ float pch[SS];
  const int nb   = nB[b];
  const int jbeg = c * SS;
  const int jend = min(jbeg + SS, nb);
  const float* pS = pScan + (size_t)b * LL;
  if (t < SS && jbeg + t < jend) pch[t] = pS[jbeg + t];
  __syncthreads();

  const float* hb = hidden + (size_t)b * LL * DD;
  v4f h = (v4f){0.f, 0.f, 0.f, 0.f};
  float Aacc = 1.0f;
  if (jbeg < jend) {
    const int npre = min(NBUF, jend - jbeg);
    for (int i = 0; i < npre; ++i)
      async_copy16((uint32_t)(uintptr_t)&xbuf[i][t],
                   hb + (size_t)(jbeg + i) * DD + 4 * t);
    for (int jj = jbeg; jj < jend; ++jj) {
      const int slot = (jj - jbeg) & (NBUF - 1);
      wait_async_rows(jend - jj);       // row jj has landed in LDS
      const v4f  x = xbuf[slot][t];
      const float p = pch[jj - jbeg];
      wait_ds0();                       // LDS reads retired -> slot reusable
      if (jj + NBUF < jend)
        async_copy16((uint32_t)(uintptr_t)&xbuf[slot][t],
                     hb + (size_t)(jj + NBUF) * DD + 4 * t);
      const float a = 1.0f - p;
      h = a * h + p * x;
      Aacc *= a;
    }
  }
  ((v4f*)(E + (size_t)(b * CC + c) * DD))[t] = h;
  if (t == 0) A[b * CC + c] = Aacc;
}

// ---------------- K3: serial chunk combine ----------------
// Hin[b,c,:] = global state entering chunk c;  H_c = E_c + A_c * H_{c-1}.
__global__ __launch_bounds__(1024)
void k3_carry(const float* __restrict__ E, const float* __restrict__ A,
              float* __restrict__ Hin) {
  const int b = blockIdx.x;
  const int d = threadIdx.x;            // 0..1023
  __shared__ float sA[CC];
  if (d < CC) sA[d] = A[b * CC + d];
  __syncthreads();
  const float* Eb = E  + (size_t)b * CC * DD + d;
  float*       Hb = Hin + (size_t)b * CC * DD + d;
  float H = 0.f;
#pragma unroll
  for (int cc = 0; cc < CC; ++cc) {
    Hb[(size_t)cc * DD] = H;
    H = Eb[(size_t)cc * DD] + sA[cc] * H;
  }
}

// ---------------- K4: pass 2 — rescan chunk with carry, scatter segments ----
__global__ __launch_bounds__(256)
void k4_pass2(const float* __restrict__ hidden, const float* __restrict__ pScan,
              const int* __restrict__ bpos, const int* __restrict__ nB,
              const float* __restrict__ Hin, float* __restrict__ out) {
  const int c = blockIdx.x, b = blockIdx.y;
  const int t = threadIdx.x;
  const int nb   = nB[b];
  const int jbeg = c * SS;
  const int jend = min(jbeg + SS, nb);
  if (jbeg >= jend) return;             // uniform per block

  __shared__ v4f  xbuf[NBUF][256];
  __shared__ float pch[SS];
  const float* pS = pScan + (size_t)b * LL;
  const int*   bp = bpos + (size_t)b * (LL + 1);
  if (t < SS && jbeg + t < jend) pch[t] = pS[jbeg + t];
  __syncthreads();

  const float* hb = hidden + (size_t)b * LL * DD;
  v4f* ob = (v4f*)out + (size_t)b * LL * (DD / 4) + t;
  v4f h = ((const v4f*)(Hin + (size_t)(b * CC + c) * DD))[t];

  const int npre = min(NBUF, jend - jbeg);
  for (int i = 0; i < npre; ++i)
    async_copy16((uint32_t)(uintptr_t)&xbuf[i][t],
                 hb + (size_t)(jbeg + i) * DD + 4 * t);

  int snext = bp[jbeg];                 // start row of first state's segment
  for (int jj = jbeg; jj < jend; ++jj) {
    const int slot = (jj - jbeg) & (NBUF - 1);
    wait_async_rows(jend - jj);
    const v4f  x = xbuf[slot][t];
    const float p = pch[jj - jbeg];
    wait_ds0();
    if (jj + NBUF < jend)
      async_copy16((uint32_t)(uintptr_t)&xbuf[slot][t],
                   hb + (size_t)(jj + NBUF) * DD + 4 * t);
    const float a = 1.0f - p;
    h = a * h + p * x;
    const int s0 = snext;
    const int e0 = bp[jj + 1];          // bpos[nb] == L sentinel
    snext = e0;
    for (int l = s0; l < e0; ++l)
      __builtin_nontemporal_store(h, &ob[(size_t)l * (DD / 4)]);
  }
}

extern "C" void kernel_launch(void* const* d_in, const int* in_sizes, int n_in,
                              void* d_out, int out_size, void* d_ws, size_t ws_size,
                              hipStream_t stream) {
  const float* hidden = (const float*)d_in[0];   // (B,L,D) f32
  const int*   bmask  = (const int*)d_in[1];     // (B,L) bool->int
  const float* bprob  = (const float*)d_in[2];   // (B,L,2) f32
  // d_in[3] = mask, all-ones, unused.
  float* out = (float*)d_out;

  // Workspace layout (all 4-byte elems; E/Hin offsets are 16B-aligned).
  float* ws    = (float*)d_ws;
  float* pScan = ws;                                   // BB*LL floats
  int*   bpos  = (int*)(ws + BB * LL);                 // BB*(LL+1) ints
  int*   nB    = (int*)(ws + BB * LL) + BB * (LL + 1); // BB ints
  float* A     = ws + BB * LL + BB * (LL + 1) + BB;    // BB*CC floats
  float* E     = A + BB * CC;                          // BB*CC*DD floats
  float* Hin   = E + (size_t)BB * CC * DD;             // BB*CC*DD floats

  k1_prep<<<dim3(BB), dim3(1024), 0, stream>>>(bmask, bprob, pScan, bpos, nB);
  k2_pass1<<<dim3(CC, BB), dim3(256), 0, stream>>>(hidden, pScan, nB, E, A);
  k3_carry<<<dim3(BB), dim3(1024), 0, stream>>>(E, A, Hin);
  k4_pass2<<<dim3(CC, BB), dim3(256), 0, stream>>>(hidden, pScan, bpos, nB, Hin, out);
}